// ATD_CA_13846974562648
// MI455X (gfx1250) — compile-verified
//
#include <hip/hip_runtime.h>

typedef float v2f __attribute__((ext_vector_type(2)));
typedef float v4f __attribute__((ext_vector_type(4)));
typedef float v8f __attribute__((ext_vector_type(8)));

#define B_     32
#define N_     4096
#define DIM_   192
#define M_     64
#define RD_    10
#define TILE_N 128
#define VSTR   200                     // LDS row stride for v: 2*200 mod 64 = 16 -> halves hit disjoint banks
#define LN_M   4.1588830833596715f     // ln(64)

static __device__ __forceinline__ v8f wmma_f32(v2f a, v2f b, v8f c) {
  // D = A(16x4,f32) * B(4x16,f32) + C(16x16,f32)
  return __builtin_amdgcn_wmma_f32_16x16x4_f32(false, a, false, b, (short)0, c,
                                               false, false);
}

// ---------------------------------------------------------------------------
// Kernel 1: per-batch precompute of normalized k (B,64,10) and v (B,64,192)
// ---------------------------------------------------------------------------
__global__ __launch_bounds__(256) void prep_kv(
    const float* __restrict__ td, const float* __restrict__ wk_w,
    const float* __restrict__ wk_b, const float* __restrict__ wv_w,
    const float* __restrict__ wv_b, float* __restrict__ kn_ws,
    float* __restrict__ v_ws) {
  const int b = blockIdx.x;
  const int tid = threadIdx.x;

  // --- k projection (scalar, tiny) ---
  __shared__ float s_k[M_ * RD_];
  for (int idx = tid; idx < M_ * RD_; idx += 256) {
    const int m = idx / RD_, d = idx - m * RD_;
    const float* tr = td + ((size_t)b * M_ + m) * DIM_;
    const float* wr = wk_w + (size_t)d * DIM_;
    float acc = wk_b[d];
    for (int c = 0; c < DIM_; c += 4)
      acc += tr[c] * wr[c] + tr[c + 1] * wr[c + 1] + tr[c + 2] * wr[c + 2] +
             tr[c + 3] * wr[c + 3];
    s_k[idx] = acc;
  }
  __syncthreads();
  if (tid < M_) {
    float n2 = 0.f;
#pragma unroll
    for (int d = 0; d < RD_; ++d) { float t = s_k[tid * RD_ + d]; n2 += t * t; }
    const float rn = 1.0f / fmaxf(sqrtf(n2), 1e-12f);
#pragma unroll
    for (int d = 0; d < RD_; ++d)
      kn_ws[((size_t)b * M_ + tid) * RD_ + d] = s_k[tid * RD_ + d] * rn;
  }

  // --- v = td @ wv_w^T + wv_b via f32 WMMA (4 x 12 tiles of 16x16, K=192) ---
  const int wid = tid >> 5, lane = tid & 31, col = lane & 15;
  const bool hi = lane >= 16;
  const int koff = hi ? 2 : 0;
  for (int tile = wid; tile < 48; tile += 8) {
    const int mt = tile / 12, et = tile - mt * 12;
    const int m0 = mt * 16, e0 = et * 16;
    const float* arow = td + ((size_t)b * M_ + m0 + col) * DIM_;
    const float* brow = wv_w + (size_t)(e0 + col) * DIM_;
    v8f c = {};
    for (int k0 = 0; k0 < DIM_; k0 += 4) {
      v2f a = *(const v2f*)(arow + k0 + koff);   // A[M=col][k0+koff, +1]
      v2f bb;
      bb.x = brow[k0 + koff];                    // B[k][e] = wv_w[e][k]
      bb.y = brow[k0 + koff + 1];
      c = wmma_f32(a, bb, c);
    }
    const float bias = wv_b[e0 + col];
    const int rsel = hi ? 8 : 0;
#pragma unroll
    for (int j = 0; j < 8; ++j)
      v_ws[((size_t)b * M_ + m0 + j + rsel) * DIM_ + e0 + col] = c[j] + bias;
  }
}

// ---------------------------------------------------------------------------
// Kernel 2: q-projection + cosine softmax + out = attn @ v
// One block = one batch x 128 rows; one wave = 16 rows.
// ---------------------------------------------------------------------------
__global__ __launch_bounds__(256) void attn_main(
    const float* __restrict__ x, const float* __restrict__ wq_w,
    const float* __restrict__ wq_b, const float* __restrict__ scale,
    const float* __restrict__ kn_ws, const float* __restrict__ v_ws,
    float* __restrict__ out, float* __restrict__ attn_out) {
  __shared__ float smem[M_ * VSTR];      // 12800 f: stage A = wqT+q, stage B = v
  __shared__ float s_kn[M_ * RD_];
  __shared__ float s_qb[RD_];
  float* s_wqT = smem;                   // [192][16], zero-padded cols 10..15
  float* s_q   = smem + DIM_ * 16;       // [8 waves][16 rows][stride 20]
  float* s_v   = smem;                   // [64][VSTR]

  const int tid = threadIdx.x;
  const int b = blockIdx.x >> 5;         // N_/TILE_N == 32 tiles
  const int rbase_blk = (blockIdx.x & 31) * TILE_N;
  const int wid = tid >> 5, lane = tid & 31, col = lane & 15;
  const bool hi = lane >= 16;
  const int koff = hi ? 2 : 0;
  const int rbase = rbase_blk + wid * 16;

  // stage-A cooperative loads
  for (int idx = tid; idx < DIM_ * 16; idx += 256) {
    const int k = idx >> 4, d = idx & 15;
    s_wqT[idx] = (d < RD_) ? wq_w[(size_t)d * DIM_ + k] : 0.0f;
  }
  for (int idx = tid; idx < M_ * RD_; idx += 256)
    s_kn[idx] = kn_ws[(size_t)b * M_ * RD_ + idx];
  if (tid < RD_) s_qb[tid] = wq_b[tid];
  __syncthreads();

  // q tile (16 x 16, cols 0..9 valid) via 48 WMMAs, K=192
  {
    const float* xrow = x + ((size_t)b * N_ + rbase + col) * DIM_;
    v8f cq = {};
    for (int k0 = 0; k0 < DIM_; k0 += 4) {
      v2f a = *(const v2f*)(xrow + k0 + koff);
      v2f bb;
      bb.x = s_wqT[(k0 + koff) * 16 + col];
      bb.y = s_wqT[(k0 + koff + 1) * 16 + col];
      cq = wmma_f32(a, bb, cq);
    }
    const int rsel = hi ? 8 : 0;
#pragma unroll
    for (int j = 0; j < 8; ++j)
      s_q[wid * 320 + (j + rsel) * 20 + col] = cq[j];   // [row][dim]
  }
  __syncthreads();

  // per-lane row: lane owns row (lane&15); halves duplicate (A-frag layout)
  float q[RD_];
#pragma unroll
  for (int d = 0; d < RD_; ++d) q[d] = s_q[wid * 320 + col * 20 + d] + s_qb[d];
  float n2 = 0.f;
#pragma unroll
  for (int d = 0; d < RD_; ++d) n2 += q[d] * q[d];
  const float rn = 1.0f / fmaxf(sqrtf(n2), 1e-12f);
#pragma unroll
  for (int d = 0; d < RD_; ++d) q[d] *= rn;

  float sc = fminf(fmaxf(scale[0], 0.0f), 3.0f);
  const float alpha = 1.0f + sc * LN_M;

  float att[M_];
#pragma unroll
  for (int m = 0; m < M_; ++m) {
    float acc = 0.f;
#pragma unroll
    for (int d = 0; d < RD_; ++d) acc += q[d] * s_kn[m * RD_ + d];
    att[m] = acc * alpha;
  }
  float mx = -3.4e38f;
#pragma unroll
  for (int m = 0; m < M_; ++m) mx = fmaxf(mx, att[m]);
  float s = 0.f;
#pragma unroll
  for (int m = 0; m < M_; ++m) { att[m] = __expf(att[m] - mx); s += att[m]; }
  const float is = 1.0f / s;
#pragma unroll
  for (int m = 0; m < M_; ++m) att[m] *= is;

  if (!hi) {  // write attn rows (low half only; high half is a duplicate)
    float* ap = attn_out + ((size_t)b * N_ + rbase + col) * M_;
#pragma unroll
    for (int m4 = 0; m4 < M_; m4 += 4) {
      v4f v; v.x = att[m4]; v.y = att[m4 + 1]; v.z = att[m4 + 2]; v.w = att[m4 + 3];
      *(v4f*)(ap + m4) = v;
    }
  }
  __syncthreads();   // all waves done with s_wqT / s_q before overwriting with v

  // stage-B: v tile -> LDS (stride VSTR for conflict-free B-fragments)
  for (int i4 = tid; i4 < (M_ * DIM_) / 4; i4 += 256) {
    const int m = i4 / 48, e4 = i4 - m * 48;
    *(v4f*)(s_v + m * VSTR + e4 * 4) =
        *(const v4f*)(v_ws + ((size_t)b * M_ + m) * DIM_ + e4 * 4);
  }
  __syncthreads();

  // out(16x192) = attn(16x64) @ v(64x192): 12 N-tiles x 16 K-steps of WMMA
  for (int nt = 0; nt < 12; ++nt) {
    const int n0 = nt * 16;
    v8f c = {};
#pragma unroll
    for (int m0 = 0; m0 < M_; m0 += 4) {
      v2f a;
      a.x = hi ? att[m0 + 2] : att[m0];
      a.y = hi ? att[m0 + 3] : att[m0 + 1];
      v2f bb;
      bb.x = s_v[(m0 + koff) * VSTR + n0 + col];
      bb.y = s_v[(m0 + koff + 1) * VSTR + n0 + col];
      c = wmma_f32(a, bb, c);
    }
    const int rsel = hi ? 8 : 0;
#pragma unroll
    for (int j = 0; j < 8; ++j)
      out[((size_t)b * N_ + rbase + j + rsel) * DIM_ + n0 + col] = c[j];
  }
}

extern "C" void kernel_launch(void* const* d_in, const int* in_sizes, int n_in,
                              void* d_out, int out_size, void* d_ws,
                              size_t ws_size, hipStream_t stream) {
  (void)in_sizes; (void)n_in; (void)out_size; (void)ws_size;
  const float* x     = (const float*)d_in[0];
  const float* td    = (const float*)d_in[1];
  const float* wq_w  = (const float*)d_in[2];
  const float* wq_b  = (const float*)d_in[3];
  const float* wk_w  = (const float*)d_in[4];
  const float* wk_b  = (const float*)d_in[5];
  const float* wv_w  = (const float*)d_in[6];
  const float* wv_b  = (const float*)d_in[7];
  const float* scale = (const float*)d_in[8];
  // d_in[9] = x_size (unused by the math)

  float* out      = (float*)d_out;
  float* attn_out = out + (size_t)B_ * N_ * DIM_;

  float* kn_ws = (float*)d_ws;          // B*64*10 floats (padded region 128KB)
  float* v_ws  = kn_ws + 32768;         // B*64*192 floats

  prep_kv<<<B_, 256, 0, stream>>>(td, wk_w, wk_b, wv_w, wv_b, kn_ws, v_ws);
  attn_main<<<B_ * (N_ / TILE_N), 256, 0, stream>>>(x, wq_w, wq_b, scale,
                                                    kn_ws, v_ws, out, attn_out);
}